// GATModel_3350074491437
// MI455X (gfx1250) — compile-verified
//
#include <hip/hip_runtime.h>
#include <hip/hip_bf16.h>
#include <math.h>

// ---------------------------------------------------------------------------
// GATv2 + GraphConv + edge MLP forward for MI455X (gfx1250, wave32).
// Dense node-feature GEMMs use V_WMMA_F32_16X16X4_F32 (fp32 WMMA).
// N = 50000 (multiple of 16), E = 800000.
// ---------------------------------------------------------------------------

#define N_NODES 50000
#define N_EDGES 800000
#define CDIV(a, b) (((a) + (b) - 1) / (b))

typedef float v2f __attribute__((ext_vector_type(2)));
typedef float v8f __attribute__((ext_vector_type(8)));

// Order-preserving float <-> int transform for atomicMax on floats.
__device__ __forceinline__ int f2o(float f) {
  int i = __float_as_int(f);
  return (i >= 0) ? i : (i ^ 0x7fffffff);
}
__device__ __forceinline__ float o2f(int i) {
  return __int_as_float((i >= 0) ? i : (i ^ 0x7fffffff));
}
__device__ __forceinline__ float elu1(float x) { return x > 0.f ? x : (expf(x) - 1.f); }
__device__ __forceinline__ float lrelu(float x) { return x > 0.f ? x : 0.2f * x; }

// ---------------------------------------------------------------------------
// Generic fill
// ---------------------------------------------------------------------------
__global__ void fill_u32(uint32_t* p, uint32_t v, int n) {
  int i = blockIdx.x * blockDim.x + threadIdx.x;
  if (i < n) p[i] = v;
}

// ---------------------------------------------------------------------------
// WMMA fp32 GEMM:  C[Nrows,Kout] = A[Nrows,Kin] @ W[Kin,Kout] (+ bias)
// One wave per 16x16 output tile, K stepped by 4 (V_WMMA_F32_16X16X4_F32).
// Nrows, Kin, Kout all multiples of 16 (Kin multiple of 4).
// ---------------------------------------------------------------------------
__global__ void gemm_wmma_f32(const float* __restrict__ A, const float* __restrict__ W,
                              const float* __restrict__ bias, float* __restrict__ C,
                              int Nrows, int Kin, int Kout) {
  const int wavesPerBlock = blockDim.x >> 5;
  const int wave = blockIdx.x * wavesPerBlock + (threadIdx.x >> 5);
  const int lane = threadIdx.x & 31;
  const int tjCount = Kout >> 4;
  const int numTiles = (Nrows >> 4) * tjCount;
  if (wave >= numTiles) return;  // wave-uniform exit: EXEC stays all-ones for WMMA
  const int ti = wave / tjCount;
  const int tj = wave - ti * tjCount;
  const int m = lane & 15;
  const int hi = lane >> 4;  // 0: K lanes {k,k+1}, 1: {k+2,k+3}
  const int row = (ti << 4) + m;
  const int col = (tj << 4) + m;
  const float* __restrict__ arow = A + (size_t)row * Kin;

  v8f acc = {};
  for (int k = 0; k < Kin; k += 4) {
    const int kk = k + hi * 2;
    v2f a, b;
    a.x = arow[kk];
    a.y = arow[kk + 1];
    b.x = W[(size_t)kk * Kout + col];
    b.y = W[(size_t)(kk + 1) * Kout + col];
    acc = __builtin_amdgcn_wmma_f32_16x16x4_f32(false, a, false, b, (short)0, acc,
                                                false, false);
  }
  const float bv = bias ? bias[col] : 0.0f;
  const int orow = (ti << 4) + hi * 8;  // D: VGPR r -> M = r (lanes 0-15) / 8+r (16-31)
#pragma unroll
  for (int r = 0; r < 8; ++r) C[(size_t)(orow + r) * Kout + col] = acc[r] + bv;
}

// ---------------------------------------------------------------------------
// Edge-weight normalization (EdgeWeightNorm 'both')
// ---------------------------------------------------------------------------
__global__ void edge_wdeg(const float* __restrict__ w, const int* __restrict__ src,
                          const int* __restrict__ dst, float* wdo, float* wdi, int E) {
  int e = blockIdx.x * blockDim.x + threadIdx.x;
  if (e >= E) return;
  float v = w[e];
  atomicAdd(&wdo[src[e]], v);
  atomicAdd(&wdi[dst[e]], v);
}

__global__ void edge_norm(const float* __restrict__ w, const int* __restrict__ src,
                          const int* __restrict__ dst, const float* __restrict__ wdo,
                          const float* __restrict__ wdi, float* nw, int E) {
  int e = blockIdx.x * blockDim.x + threadIdx.x;
  if (e >= E) return;
  nw[e] = w[e] * rsqrtf(wdo[src[e]]) * rsqrtf(wdi[dst[e]]);
}

// ---------------------------------------------------------------------------
// GATv2 attention: pass 1 — logits + segment max (via ordered-int atomicMax)
// ---------------------------------------------------------------------------
__global__ void att_logits(const float* __restrict__ fs, const float* __restrict__ fd,
                           const float* __restrict__ attn, const int* __restrict__ src,
                           const int* __restrict__ dst, float* __restrict__ elog,
                           int* __restrict__ mbuf, int E, int H, int D) {
  int e = blockIdx.x * blockDim.x + threadIdx.x;
  if (e >= E) return;
  const int s = src[e], t = dst[e];
  const float* ps = fs + (size_t)s * H * D;
  const float* pt = fd + (size_t)t * H * D;
  for (int h = 0; h < H; ++h) {
    float acc = 0.f;
    for (int d = 0; d < D; ++d) {
      float v = ps[h * D + d] + pt[h * D + d];
      acc += lrelu(v) * attn[h * D + d];
    }
    elog[(size_t)e * H + h] = acc;
    atomicMax(&mbuf[t * H + h], f2o(acc));
  }
}

// pass 2 — exp(logit - max) + segment sum
__global__ void att_expsum(const int* __restrict__ dst, float* __restrict__ elog,
                           const int* __restrict__ mbuf, float* __restrict__ sbuf,
                           int E, int H) {
  int idx = blockIdx.x * blockDim.x + threadIdx.x;
  if (idx >= E * H) return;
  int e = idx / H;
  int h = idx - e * H;
  int t = dst[e];
  float num = expf(elog[idx] - o2f(mbuf[t * H + h]));
  elog[idx] = num;
  atomicAdd(&sbuf[t * H + h], num);
}

// pass 3 — alpha-weighted scatter of fs[src]
__global__ void att_aggregate(const float* __restrict__ fs, const int* __restrict__ src,
                              const int* __restrict__ dst, const float* __restrict__ elog,
                              const float* __restrict__ sbuf, float* __restrict__ agg,
                              int E, int H, int D) {
  int e = blockIdx.x * blockDim.x + threadIdx.x;
  if (e >= E) return;
  const int s = src[e], t = dst[e];
  for (int h = 0; h < H; ++h) {
    float alpha = elog[(size_t)e * H + h] / sbuf[t * H + h];
    const float* ps = fs + (size_t)s * H * D + (size_t)h * D;
    float* pa = agg + (size_t)t * H * D + (size_t)h * D;
    for (int d = 0; d < D; ++d) atomicAdd(&pa[d], ps[d] * alpha);
  }
}

// ---------------------------------------------------------------------------
// Elementwise helpers
// ---------------------------------------------------------------------------
__global__ void elu_map(const float* __restrict__ in, float* __restrict__ out, int n) {
  int i = blockIdx.x * blockDim.x + threadIdx.x;
  if (i < n) out[i] = elu1(in[i]);
}

__global__ void bias_elu(const float* __restrict__ in, const float* __restrict__ bias,
                         float* __restrict__ out, int total, int D) {
  int i = blockIdx.x * blockDim.x + threadIdx.x;
  if (i >= total) return;
  out[i] = elu1(in[i] + bias[i % D]);
}

// GraphConv (norm='none', weighted) edge scatter
__global__ void gc_weighted_agg(const float* __restrict__ xm, const int* __restrict__ src,
                                const int* __restrict__ dst, const float* __restrict__ nw,
                                float* __restrict__ agg, int E, int D) {
  int e = blockIdx.x * blockDim.x + threadIdx.x;
  if (e >= E) return;
  float w = nw[e];
  const float* ps = xm + (size_t)src[e] * D;
  float* pa = agg + (size_t)dst[e] * D;
  for (int d = 0; d < D; ++d) atomicAdd(&pa[d], ps[d] * w);
}

// unweighted degree counts
__global__ void edge_count(const int* __restrict__ src, const int* __restrict__ dst,
                           float* dout, float* din, int E) {
  int e = blockIdx.x * blockDim.x + threadIdx.x;
  if (e >= E) return;
  atomicAdd(&dout[src[e]], 1.0f);
  atomicAdd(&din[dst[e]], 1.0f);
}

// x[n,:] *= rsqrt(max(deg[n],1))
__global__ void scale_rsqrt(float* __restrict__ x, const float* __restrict__ deg,
                            int total, int D) {
  int i = blockIdx.x * blockDim.x + threadIdx.x;
  if (i >= total) return;
  x[i] *= rsqrtf(fmaxf(deg[i / D], 1.0f));
}

// GraphConv (norm='both') edge scatter, unweighted
__global__ void gc_agg(const float* __restrict__ xs, const int* __restrict__ src,
                       const int* __restrict__ dst, float* __restrict__ agg, int E, int D) {
  int e = blockIdx.x * blockDim.x + threadIdx.x;
  if (e >= E) return;
  const float* ps = xs + (size_t)src[e] * D;
  float* pa = agg + (size_t)dst[e] * D;
  for (int d = 0; d < D; ++d) atomicAdd(&pa[d], ps[d]);
}

// out = elu(agg * rsqrt(max(din,1)) + bias)
__global__ void gc2_finish(const float* __restrict__ agg, const float* __restrict__ din,
                           const float* __restrict__ bias, float* __restrict__ out,
                           int total, int D) {
  int i = blockIdx.x * blockDim.x + threadIdx.x;
  if (i >= total) return;
  int n = i / D;
  int d = i - n * D;
  out[i] = elu1(agg[i] * rsqrtf(fmaxf(din[n], 1.0f)) + bias[d]);
}

// Edge classifier: score[e,c] = concat(x4[src], x4[dst]) @ Wcls + bcls
__global__ void edge_cls(const float* __restrict__ x, const int* __restrict__ src,
                         const int* __restrict__ dst, const float* __restrict__ Wc,
                         const float* __restrict__ bc, float* __restrict__ out, int E) {
  int e = blockIdx.x * blockDim.x + threadIdx.x;
  if (e >= E) return;
  const float* xs = x + (size_t)src[e] * 16;
  const float* xd = x + (size_t)dst[e] * 16;
  float acc[8];
#pragma unroll
  for (int c = 0; c < 8; ++c) acc[c] = bc[c];
#pragma unroll
  for (int k = 0; k < 16; ++k) {
    float v = xs[k];
#pragma unroll
    for (int c = 0; c < 8; ++c) acc[c] += v * Wc[k * 8 + c];
  }
#pragma unroll
  for (int k = 0; k < 16; ++k) {
    float v = xd[k];
#pragma unroll
    for (int c = 0; c < 8; ++c) acc[c] += v * Wc[(16 + k) * 8 + c];
  }
#pragma unroll
  for (int c = 0; c < 8; ++c) out[(size_t)e * 8 + c] = acc[c];
}

// ---------------------------------------------------------------------------
// Host-side orchestration
// ---------------------------------------------------------------------------
static inline void launch_gemm(const float* A, const float* W, const float* bias, float* C,
                               int Nrows, int Kin, int Kout, hipStream_t stream) {
  int waves = (Nrows >> 4) * (Kout >> 4);
  int blocks = CDIV(waves, 8);  // 256 threads = 8 waves per block
  gemm_wmma_f32<<<blocks, 256, 0, stream>>>(A, W, bias, C, Nrows, Kin, Kout);
}

extern "C" void kernel_launch(void* const* d_in, const int* in_sizes, int n_in,
                              void* d_out, int out_size, void* d_ws, size_t ws_size,
                              hipStream_t stream) {
  (void)in_sizes; (void)n_in; (void)out_size; (void)ws_size;
  const int N = N_NODES, E = N_EDGES;
  const float* h     = (const float*)d_in[0];
  const float* edgew = (const float*)d_in[1];
  const int*   src   = (const int*)d_in[2];
  const int*   dst   = (const int*)d_in[3];
  const float* Wn    = (const float*)d_in[4];
  const float* bn    = (const float*)d_in[5];
  const float* W1s   = (const float*)d_in[6];
  const float* b1s   = (const float*)d_in[7];
  const float* W1d   = (const float*)d_in[8];
  const float* b1d   = (const float*)d_in[9];
  const float* attn1 = (const float*)d_in[10];
  const float* W2s   = (const float*)d_in[11];
  const float* b2s   = (const float*)d_in[12];
  const float* W2d   = (const float*)d_in[13];
  const float* b2d   = (const float*)d_in[14];
  const float* attn2 = (const float*)d_in[15];
  const float* Wc1   = (const float*)d_in[16];
  const float* bc1   = (const float*)d_in[17];
  const float* Wc2   = (const float*)d_in[18];
  const float* bc2   = (const float*)d_in[19];
  const float* Wcls  = (const float*)d_in[20];
  const float* bcls  = (const float*)d_in[21];
  float* out = (float*)d_out;

  // Workspace carve-up (floats)
  float* ws    = (float*)d_ws;
  float* nf    = ws;           ws += (size_t)N * 128;  // node_f
  float* fs    = ws;           ws += (size_t)N * 64;   // src projection (both layers)
  float* fd    = ws;           ws += (size_t)N * 64;   // dst projection (both layers)
  float* xcur  = ws;           ws += (size_t)N * 64;   // current node features
  float* agg   = ws;           ws += (size_t)N * 64;   // aggregation target (64/32/16)
  int*   mbuf  = (int*)ws;     ws += (size_t)N * 2;    // segment max (ordered-int)
  float* sbuf  = ws;           ws += (size_t)N * 2;    // segment sum
  float* elog  = ws;           ws += (size_t)E * 2;    // logits -> exp numerators
  float* normw = ws;           ws += (size_t)E;        // normalized edge weights
  float* wdo   = ws;           ws += (size_t)N;        // out degree (weighted, then count)
  float* wdi   = ws;           ws += (size_t)N;        // in degree
  float* xm32  = ws;           ws += (size_t)N * 32;   // GraphConv1 projection
  float* xs16  = ws;           ws += (size_t)N * 16;   // GraphConv2 projection / x4

  const int T = 256;
  const uint32_t NEG_INF_ORD = 0x807fffffu;  // f2o(-inf)

  // --- EdgeWeightNorm('both') ---
  fill_u32<<<CDIV(2 * N, T), T, 0, stream>>>((uint32_t*)wdo, 0u, 2 * N);  // wdo|wdi contiguous
  edge_wdeg<<<CDIV(E, T), T, 0, stream>>>(edgew, src, dst, wdo, wdi, E);
  edge_norm<<<CDIV(E, T), T, 0, stream>>>(edgew, src, dst, wdo, wdi, normw, E);

  // --- node_f = h @ Wn + bn ---
  launch_gemm(h, Wn, bn, nf, N, 128, 128, stream);

  // --- GATv2 layer 1 (H=2, D=32) ---
  launch_gemm(nf, W1s, b1s, fs, N, 128, 64, stream);
  launch_gemm(nf, W1d, b1d, fd, N, 128, 64, stream);
  fill_u32<<<CDIV(2 * N, T), T, 0, stream>>>((uint32_t*)mbuf, NEG_INF_ORD, 2 * N);
  fill_u32<<<CDIV(2 * N, T), T, 0, stream>>>((uint32_t*)sbuf, 0u, 2 * N);
  fill_u32<<<CDIV(N * 64, T), T, 0, stream>>>((uint32_t*)agg, 0u, N * 64);
  att_logits<<<CDIV(E, T), T, 0, stream>>>(fs, fd, attn1, src, dst, elog, mbuf, E, 2, 32);
  att_expsum<<<CDIV(E * 2, T), T, 0, stream>>>(dst, elog, mbuf, sbuf, E, 2);
  att_aggregate<<<CDIV(E, T), T, 0, stream>>>(fs, src, dst, elog, sbuf, agg, E, 2, 32);
  elu_map<<<CDIV(N * 64, T), T, 0, stream>>>(agg, xcur, N * 64);

  // --- GATv2 layer 2 (H=1, D=64) ---
  launch_gemm(xcur, W2s, b2s, fs, N, 64, 64, stream);
  launch_gemm(xcur, W2d, b2d, fd, N, 64, 64, stream);
  fill_u32<<<CDIV(N, T), T, 0, stream>>>((uint32_t*)mbuf, NEG_INF_ORD, N);
  fill_u32<<<CDIV(N, T), T, 0, stream>>>((uint32_t*)sbuf, 0u, N);
  fill_u32<<<CDIV(N * 64, T), T, 0, stream>>>((uint32_t*)agg, 0u, N * 64);
  att_logits<<<CDIV(E, T), T, 0, stream>>>(fs, fd, attn2, src, dst, elog, mbuf, E, 1, 64);
  att_expsum<<<CDIV(E, T), T, 0, stream>>>(dst, elog, mbuf, sbuf, E, 1);
  att_aggregate<<<CDIV(E, T), T, 0, stream>>>(fs, src, dst, elog, sbuf, agg, E, 1, 64);
  elu_map<<<CDIV(N * 64, T), T, 0, stream>>>(agg, xcur, N * 64);

  // --- GraphConv1 (norm='none', weighted): elu(segsum(norm_w * (x@Wc1)[src]) + bc1) ---
  launch_gemm(xcur, Wc1, nullptr, xm32, N, 64, 32, stream);
  fill_u32<<<CDIV(N * 32, T), T, 0, stream>>>((uint32_t*)agg, 0u, N * 32);
  gc_weighted_agg<<<CDIV(E, T), T, 0, stream>>>(xm32, src, dst, normw, agg, E, 32);
  bias_elu<<<CDIV(N * 32, T), T, 0, stream>>>(agg, bc1, xcur, N * 32, 32);  // x3 in xcur[N*32]

  // --- GraphConv2 (norm='both', unweighted degrees clamped to >=1) ---
  fill_u32<<<CDIV(2 * N, T), T, 0, stream>>>((uint32_t*)wdo, 0u, 2 * N);
  edge_count<<<CDIV(E, T), T, 0, stream>>>(src, dst, wdo, wdi, E);
  launch_gemm(xcur, Wc2, nullptr, xs16, N, 32, 16, stream);
  scale_rsqrt<<<CDIV(N * 16, T), T, 0, stream>>>(xs16, wdo, N * 16, 16);
  fill_u32<<<CDIV(N * 16, T), T, 0, stream>>>((uint32_t*)agg, 0u, N * 16);
  gc_agg<<<CDIV(E, T), T, 0, stream>>>(xs16, src, dst, agg, E, 16);
  gc2_finish<<<CDIV(N * 16, T), T, 0, stream>>>(agg, wdi, bc2, xs16, N * 16, 16);  // x4 in xs16

  // --- Edge MLP predictor ---
  edge_cls<<<CDIV(E, T), T, 0, stream>>>(xs16, src, dst, Wcls, bcls, out, E);
}